// adj_weight_51960514347735
// MI455X (gfx1250) — compile-verified
//
#include <hip/hip_runtime.h>
#include <hip/hip_bf16.h>
#include <math.h>

typedef __attribute__((ext_vector_type(2))) float v2f;
typedef __attribute__((ext_vector_type(4))) float v4f;
typedef __attribute__((ext_vector_type(8))) float v8f;

#define N_PTS   8192
#define DIM     16
#define KNN     8
#define WAVES   8              // waves per block (column-loop split factor)
#define IDXMASK 8191u          // 13 bits for column index packed into key
#define KEYMASK 0xFFFFE000u

// ---------------------------------------------------------------------------
// K1: per-row squared norms; thread 0 zeroes the global distance-sum slot.
// ---------------------------------------------------------------------------
__global__ void sqnorm_kernel(const float* __restrict__ x,
                              float* __restrict__ sq,
                              float* __restrict__ sum_slot, int n) {
    int i = blockIdx.x * blockDim.x + threadIdx.x;
    if (i == 0) *sum_slot = 0.0f;
    if (i >= n) return;
    const float* r = x + i * DIM;
    float s = 0.0f;
#pragma unroll
    for (int d = 0; d < DIM; ++d) s += r[d] * r[d];
    sq[i] = s;
}

// ---------------------------------------------------------------------------
// K2: 8 waves per 16-row tile; wave w covers column tiles jt ≡ w (mod 8).
// Gram tiles via 4 chained v_wmma_f32_16x16x4_f32 (full f32, K=4*4=DIM).
// Each lane owns 8 (row, col-class) top-8 lists of packed keys:
//   key = (bits(dist2) & 0xFFFFE000) | col   (dist2 >= 0 -> uint-monotone)
// Merge: per-wave 128->8 per row, then wave 0 merges 8*8=64 -> 8 per row.
// ---------------------------------------------------------------------------
__global__ void __launch_bounds__(32 * WAVES)
knn_kernel(const float* __restrict__ x, const float* __restrict__ sq,
           float* __restrict__ d_knn, int* __restrict__ idx_out,
           float* __restrict__ sum_slot) {
    __shared__ unsigned int lds1[WAVES * 16 * 16 * KNN]; // 64 KB: [wave][row][laneCol][slot]
    __shared__ unsigned int lds2[16 * WAVES * KNN];      //  4 KB: [row][wave][slot]

    const int tid  = threadIdx.x;
    const int wv   = tid >> 5;
    const int lane = tid & 31;
    const int half = lane >> 4;      // 0 or 1
    const int l15  = lane & 15;
    const int i0   = blockIdx.x * 16;

    // ---- A fragments for this row tile (reused across all column tiles) ----
    v2f afrag[4];
#pragma unroll
    for (int kc = 0; kc < 4; ++kc)
        afrag[kc] = *reinterpret_cast<const v2f*>(
            x + (i0 + l15) * DIM + kc * 4 + half * 2);

    // row norms for the 8 C-rows this lane touches (M = v + 8*half)
    float sqi[8];
#pragma unroll
    for (int v = 0; v < 8; ++v) sqi[v] = sq[i0 + v + 8 * half];

    // top-8 lists (unsorted) + cached max, per C-register v
    unsigned int keys[8][KNN];
    unsigned int mx[8];
#pragma unroll
    for (int v = 0; v < 8; ++v) {
        mx[v] = 0xFFFFFFFFu;
#pragma unroll
        for (int s = 0; s < KNN; ++s) keys[v][s] = 0xFFFFFFFFu;
    }

    for (int jt = wv; jt < N_PTS / 16; jt += WAVES) {
        const int j0 = jt * 16;

        v2f bfrag[4];
#pragma unroll
        for (int kc = 0; kc < 4; ++kc)
            bfrag[kc] = *reinterpret_cast<const v2f*>(
                x + (j0 + l15) * DIM + kc * 4 + half * 2);

        v8f c = {};
#pragma unroll
        for (int kc = 0; kc < 4; ++kc)
            c = __builtin_amdgcn_wmma_f32_16x16x4_f32(
                    false, afrag[kc], false, bfrag[kc],
                    (short)0, c, false, false);

        const float sqj = sq[j0 + l15];
        const int   col = j0 + l15;

#pragma unroll
        for (int v = 0; v < 8; ++v) {
            float d = sqi[v] + sqj - 2.0f * c[v];
            d = fmaxf(d, 0.0f);
            const int row = i0 + v + 8 * half;
            unsigned int key = (row == col)
                ? 0xFFFFFFFFu
                : ((__float_as_uint(d) & KEYMASK) | (unsigned int)col);
            if (key < mx[v]) {
                bool rep = false;
#pragma unroll
                for (int s = 0; s < KNN; ++s)
                    if (!rep && keys[v][s] == mx[v]) { keys[v][s] = key; rep = true; }
                unsigned int m = keys[v][0];
#pragma unroll
                for (int s = 1; s < KNN; ++s) m = (keys[v][s] > m) ? keys[v][s] : m;
                mx[v] = m;
            }
        }
    }

    // ---- stage 1: dump per-lane lists ----
    unsigned int* seg1 = lds1 + wv * (16 * 16 * KNN);
#pragma unroll
    for (int v = 0; v < 8; ++v)
#pragma unroll
        for (int s = 0; s < KNN; ++s)
            seg1[(((v + 8 * half) * 16) + l15) * KNN + s] = keys[v][s];
    __syncthreads();

    // ---- stage 2: each wave reduces its own 128 candidates/row -> 8 ----
    if (lane < 16) {
        const int row = lane;
        unsigned int* seg = seg1 + row * 16 * KNN;   // this wave's 128 cands
        for (int s = 0; s < KNN; ++s) {
            unsigned int best = 0xFFFFFFFFu; int bi = 0;
            for (int t = 0; t < 16 * KNN; ++t) {
                unsigned int kk = seg[t];
                if (kk < best) { best = kk; bi = t; }
            }
            seg[bi] = 0xFFFFFFFFu;
            lds2[(row * WAVES + wv) * KNN + s] = best;
        }
    }
    __syncthreads();

    // ---- stage 3: wave 0 merges 8 waves * 8 = 64 candidates/row -> 8 ----
    if (tid < 16) {
        const int row = tid;
        unsigned int* seg = &lds2[row * WAVES * KNN]; // 64 candidates
        float local_sum = 0.0f;
        for (int s = 0; s < KNN; ++s) {
            unsigned int best = 0xFFFFFFFFu; int bi = 0;
            for (int t = 0; t < WAVES * KNN; ++t) {
                unsigned int kk = seg[t];
                if (kk < best) { best = kk; bi = t; }
            }
            seg[bi] = 0xFFFFFFFFu;
            float d = __uint_as_float(best & KEYMASK);
            d_knn[(i0 + row) * KNN + s]   = d;
            idx_out[(i0 + row) * KNN + s] = (int)(best & IDXMASK);
            local_sum += d;
        }
        atomicAdd(sum_slot, local_sum);
    }
}

// ---------------------------------------------------------------------------
// K3: stream-zero the 256 MB output (the bandwidth roofline term).
// ---------------------------------------------------------------------------
__global__ void zero_kernel(v4f* __restrict__ out, long long n4) {
    long long i = (long long)blockIdx.x * blockDim.x + threadIdx.x;
    if (i < n4) {
        v4f z = {0.f, 0.f, 0.f, 0.f};
        __builtin_nontemporal_store(z, out + i);
    }
}

// ---------------------------------------------------------------------------
// K4: sparse scatter of the symmetrized Laplacian.
// For directed edge (i,j,w): L[i][j]-=w/2, L[j][i]-=w/2, L[i][i]+=w/2, L[j][j]+=w/2.
// ---------------------------------------------------------------------------
__global__ void scatter_kernel(const float* __restrict__ d_knn,
                               const int* __restrict__ idx,
                               const float* __restrict__ sum_slot,
                               float* __restrict__ out, int n) {
    int e = blockIdx.x * blockDim.x + threadIdx.x;
    if (e >= n * KNN) return;
    const float t = (*sum_slot) / (float)(n * KNN) + 1e-12f;
    const int i = e >> 3;
    const int j = idx[e];
    const float hw = 0.5f * __expf(-d_knn[e] / t);
    atomicAdd(out + (long long)i * n + j, -hw);
    atomicAdd(out + (long long)j * n + i, -hw);
    atomicAdd(out + (long long)i * n + i,  hw);
    atomicAdd(out + (long long)j * n + j,  hw);
}

// ---------------------------------------------------------------------------
extern "C" void kernel_launch(void* const* d_in, const int* in_sizes, int n_in,
                              void* d_out, int out_size, void* d_ws, size_t ws_size,
                              hipStream_t stream) {
    const float* x = (const float*)d_in[0];
    const int n = in_sizes[0] / DIM;              // 8192

    // workspace layout
    char* ws = (char*)d_ws;
    float* sq       = (float*)ws;                              // n floats
    float* d_knn    = (float*)(ws + (size_t)n * 4);            // n*8 floats
    int*   idx      = (int*)  (ws + (size_t)n * 4 + (size_t)n * KNN * 4);
    float* sum_slot = (float*)(ws + (size_t)n * 4 + (size_t)n * KNN * 8);
    float* out = (float*)d_out;

    sqnorm_kernel<<<(n + 255) / 256, 256, 0, stream>>>(x, sq, sum_slot, n);
    knn_kernel<<<n / 16, 32 * WAVES, 0, stream>>>(x, sq, d_knn, idx, sum_slot);

    long long n4 = ((long long)n * n) / 4;
    zero_kernel<<<(unsigned)((n4 + 255) / 256), 256, 0, stream>>>((v4f*)out, n4);
    scatter_kernel<<<(n * KNN + 255) / 256, 256, 0, stream>>>(d_knn, idx, sum_slot, out, n);
}